// Encoder_13443247637090
// MI455X (gfx1250) — compile-verified
//
#include <hip/hip_runtime.h>

typedef __attribute__((ext_vector_type(2))) float v2f;
typedef __attribute__((ext_vector_type(4))) float v4f;
typedef __attribute__((ext_vector_type(8))) float v8f;

#define FEAT       512
#define EMBED      128
#define BATCH      16384
#define NSAMP      10
#define NT         16                 // batch columns per workgroup
#define KDIM       (2 * FEAT)         // 1024
#define LDS_STRIDE (KDIM + 4)         // +16B pad -> rotate 4 banks/row, conflict-free ds_load_b64

__global__ __launch_bounds__(256) void sage_encoder_wmma(
    const int*   __restrict__ nodes,    // [BATCH]
    const int*   __restrict__ neigh,    // [BATCH][NSAMP]
    const float* __restrict__ feat,     // [NUM_NODES][FEAT]
    const float* __restrict__ weight,   // [EMBED][KDIM]
    float*       __restrict__ out)      // [EMBED][BATCH]
{
    __shared__ float comb[NT][LDS_STRIDE];   // combined tile: [batch-in-tile][2*FEAT]

    const int tid      = threadIdx.x;
    const int lane     = tid & 31;
    const int wid      = tid >> 5;           // wave id 0..7
    const int tileBase = blockIdx.x * NT;

    // ---------------- Phase 1: fused gather + neighbor mean into LDS ----------------
    // wave w builds batch rows 2w and 2w+1 of the tile
    #pragma unroll
    for (int rr = 0; rr < 2; ++rr) {
        const int r = wid * 2 + rr;
        const int b = tileBase + r;

        // self features -> comb[r][0:512]
        const float* selfRow = feat + (long)nodes[b] * FEAT;
        #pragma unroll
        for (int j = 0; j < 4; ++j) {
            const int f4 = lane + 32 * j;                       // float4 index 0..127
            v4f v = *(const v4f*)(selfRow + 4 * f4);            // global_load_b128
            *(v4f*)(&comb[r][4 * f4]) = v;                      // ds_store_b128
        }

        // mean of 10 neighbor rows -> comb[r][512:1024]
        v4f acc[4] = {};
        #pragma unroll
        for (int k = 0; k < NSAMP; ++k) {
            const float* nRow = feat + (long)neigh[b * NSAMP + k] * FEAT;
            #pragma unroll
            for (int j = 0; j < 4; ++j) {
                v4f v = *(const v4f*)(nRow + 4 * (lane + 32 * j));
                acc[j] += v;
            }
        }
        const float inv = 1.0f / (float)NSAMP;
        #pragma unroll
        for (int j = 0; j < 4; ++j) {
            v4f m = acc[j] * inv;
            *(v4f*)(&comb[r][FEAT + 4 * (lane + 32 * j)]) = m;
        }
    }

    __syncthreads();

    // ---------------- Phase 2: WMMA f32 GEMM, wave w -> out rows [16w,16w+16) ----------------
    // A (16x4 f32):  lane&15 = M row;  VGPR0/1 hold K = {0,1} (lanes 0-15) or {2,3} (lanes 16-31)
    // B (4x16 f32):  lane&15 = N col;  same half-lane K split
    const int  mBase = wid * 16;
    const int  half  = lane >> 4;        // 0: K offset 0,1   1: K offset 2,3
    const int  l15   = lane & 15;

    const float* wRow = weight + (long)(mBase + l15) * KDIM;   // A-row for this lane
    v8f c = {};

    #pragma unroll 8
    for (int k0 = 0; k0 < KDIM; k0 += 4) {
        v2f a = *(const v2f*)(wRow + k0 + 2 * half);                 // global_load_b64 (L2-hot)
        v2f bm = *(const v2f*)(&comb[l15][k0 + 2 * half]);           // ds_load_b64, conflict-free
        c = __builtin_amdgcn_wmma_f32_16x16x4_f32(
                /*neg_a=*/false, a, /*neg_b=*/false, bm,
                /*c_mod=*/(short)0, c, /*reuse_a=*/false, /*reuse_b=*/false);
    }

    // D layout: VGPR r -> M = mBase + r + 8*half, N = tileBase + l15
    float* outBase = out + (long)(mBase + 8 * half) * BATCH + tileBase + l15;
    #pragma unroll
    for (int r = 0; r < 8; ++r) {
        float v = c[r];
        v = v > 0.0f ? v : 0.0f;                                     // ReLU
        outBase[(long)r * BATCH] = v;                                // coalesced across l15
    }
}

extern "C" void kernel_launch(void* const* d_in, const int* in_sizes, int n_in,
                              void* d_out, int out_size, void* d_ws, size_t ws_size,
                              hipStream_t stream) {
    const int*   nodes  = (const int*)d_in[0];
    const int*   neigh  = (const int*)d_in[1];
    const float* feat   = (const float*)d_in[2];
    const float* weight = (const float*)d_in[3];
    float*       out    = (float*)d_out;

    dim3 grid(BATCH / NT);   // 1024 workgroups
    dim3 block(256);         // 8 wave32 per workgroup
    sage_encoder_wmma<<<grid, block, 0, stream>>>(nodes, neigh, feat, weight, out);
}